// SimpleGraphConv_12068858102168
// MI455X (gfx1250) — compile-verified
//
#include <hip/hip_runtime.h>

#define N_NODES  50000
#define N_EDGES  800000
#define N_FEAT   64
#define N_LAYERS 3
#define N_TILES  (N_NODES / 16)   // 3125, exact

// padded stride (in float2 units) for the interleaved B tile: dword offset of
// consecutive k2-rows is 160 ≡ 32 (mod 64), so the two half-waves (k2 vs k2+1)
// land on disjoint halves of the 64 LDS banks -> conflict-free ds_load_b64.
#define SB_STRIDE2 80

typedef float v2f __attribute__((ext_vector_type(2)));
typedef float v8f __attribute__((ext_vector_type(8)));
typedef int   v4i __attribute__((ext_vector_type(4)));

typedef __attribute__((address_space(1))) v4i* gv4i_p;   // global int4*
typedef __attribute__((address_space(3))) v4i* lv4i_p;   // LDS int4*

#if defined(__HIP_DEVICE_COMPILE__) && __has_builtin(__builtin_amdgcn_global_load_async_to_lds_b128)
#define HAVE_ASYNC_LDS 1
#else
#define HAVE_ASYNC_LDS 0
#endif

// ---------------- degree kernels ----------------

__global__ void init_deg_kernel(float* __restrict__ deg) {
    int i = blockIdx.x * blockDim.x + threadIdx.x;
    if (i < N_NODES) deg[i] = 1.0f;              // self-loop contributes 1
}

__global__ void count_deg_kernel(const int* __restrict__ dst, float* __restrict__ deg) {
    int e = blockIdx.x * blockDim.x + threadIdx.x;
    if (e < N_EDGES) atomicAdd(&deg[dst[e]], 1.0f);
}

__global__ void invert_deg_kernel(float* __restrict__ deg) {
    int i = blockIdx.x * blockDim.x + threadIdx.x;
    if (i < N_NODES) deg[i] = 1.0f / deg[i];     // deg >= 1 guaranteed
}

// ---------------- aggregation ----------------

// agg = x  (self-loop term; fully overwrites agg so no zero-init pass needed)
__global__ void self_copy_kernel(const float* __restrict__ x, float* __restrict__ agg) {
    int i = blockIdx.x * blockDim.x + threadIdx.x;
    if (i < N_NODES * N_FEAT / 4)
        ((float4*)agg)[i] = ((const float4*)x)[i];
}

// 16 threads per edge, one float4 chunk each: coalesced 256B row gather,
// fp32 atomic adds resolved at L2 (working set fits 192MB L2).
__global__ void scatter_kernel(const float* __restrict__ x,
                               const int* __restrict__ src,
                               const int* __restrict__ dst,
                               float* __restrict__ agg) {
    unsigned t = blockIdx.x * blockDim.x + threadIdx.x;
    if (t >= (unsigned)N_EDGES * 16u) return;
    unsigned e = t >> 4;
    unsigned c = t & 15u;
    if (c == 0) {
        // stream-ahead prefetch of the edge index arrays (global_prefetch_b8)
        __builtin_prefetch(src + e + 2048, 0, 0);
        __builtin_prefetch(dst + e + 2048, 0, 0);
    }
    int s = src[e];
    int d = dst[e];
    float4 v = *(const float4*)(x + (size_t)s * N_FEAT + c * 4);
    float* p = agg + (size_t)d * N_FEAT + c * 4;
    atomicAdd(p + 0, v.x);
    atomicAdd(p + 1, v.y);
    atomicAdd(p + 2, v.z);
    atomicAdd(p + 3, v.w);
}

// ---------------- WMMA GEMM: out = (agg * inv_deg) @ W^T ----------------
// One wave handles a 16-row node tile (16x64 output) via fp32 WMMA 16x16x4.
// B = W^T. We stage B in LDS k-pair-interleaved: sB[k2*STRIDE + j] holds
// (B[2k2][j], B[2k2+1][j]) = (W[j][2k2], W[j][2k2+1]) -> each lane's B
// fragment is one aligned ds_load_b64, no register marshalling before WMMA.
__global__ void gemm_wmma_kernel(const float* __restrict__ agg,
                                 const float* __restrict__ invdeg,
                                 const float* __restrict__ W,
                                 float* __restrict__ out) {
#if HAVE_ASYNC_LDS
    __shared__ float sStage[N_FEAT * N_FEAT];
#endif
    __shared__ v2f sB[32 * SB_STRIDE2];

#if HAVE_ASYNC_LDS
    // Async DMA the 16KB weight matrix straight into LDS (no VGPR round-trip),
    // tracked by ASYNCcnt.
    for (int i = threadIdx.x * 4; i < N_FEAT * N_FEAT; i += blockDim.x * 4) {
        __builtin_amdgcn_global_load_async_to_lds_b128(
            (gv4i_p)(W + i),
            (lv4i_p)(sStage + i),
            0, 0);
    }
#if __has_builtin(__builtin_amdgcn_s_wait_asynccnt)
    __builtin_amdgcn_s_wait_asynccnt(0);
#else
    asm volatile("s_wait_asynccnt 0" ::: "memory");
#endif
    __syncthreads();
    const float* Wsrc = sStage;
#else
    const float* Wsrc = W;
#endif

    // interleave/transpose into the WMMA-friendly layout
    for (int i = threadIdx.x; i < N_FEAT * 32; i += blockDim.x) {
        int j  = i >> 5;        // W row  = output feature = B column
        int k2 = i & 31;        // k pair index
        sB[k2 * SB_STRIDE2 + j] = *(const v2f*)(Wsrc + j * N_FEAT + k2 * 2);
    }
    __syncthreads();

    int wave = threadIdx.x >> 5;
    int lane = threadIdx.x & 31;
    int tile = blockIdx.x * (blockDim.x >> 5) + wave;
    if (tile >= N_TILES) return;                  // wave-uniform: EXEC all-ones for WMMA

    int m0    = tile * 16;
    int n     = lane & 15;                        // N index of this lane (B/C/D layout)
    int khalf = lane >> 4;                        // A/B: lanes 16-31 hold K+2/K+3
    int row   = m0 + n;                           // A layout: lane%16 = M row
    float scale = invdeg[row];
    const float* arow = agg + (size_t)row * N_FEAT + 2 * khalf;
    const v2f*   Brow = sB + khalf * SB_STRIDE2;  // lanes 16-31 take the next k2-row

    v8f acc0 = {}, acc1 = {}, acc2 = {}, acc3 = {};

#pragma unroll
    for (int k0 = 0; k0 < N_FEAT; k0 += 4) {
        v2f a = *(const v2f*)(arow + k0);         // A[row][kb], A[row][kb+1]
        a = a * scale;                            // v_pk_mul_f32

        const v2f* bp = Brow + (k0 >> 1) * SB_STRIDE2 + n;
        v2f b0 = bp[0];                           // columns n, n+16, n+32, n+48
        v2f b1 = bp[16];
        v2f b2 = bp[32];
        v2f b3 = bp[48];

        acc0 = __builtin_amdgcn_wmma_f32_16x16x4_f32(false, a, false, b0, (short)0, acc0, false, false);
        acc1 = __builtin_amdgcn_wmma_f32_16x16x4_f32(false, a, false, b1, (short)0, acc1, false, false);
        acc2 = __builtin_amdgcn_wmma_f32_16x16x4_f32(false, a, false, b2, (short)0, acc2, false, false);
        acc3 = __builtin_amdgcn_wmma_f32_16x16x4_f32(false, a, false, b3, (short)0, acc3, false, false);
    }

    // C/D layout: VGPR v -> M = v (+8 for lanes 16-31), N = lane&15
#pragma unroll
    for (int v = 0; v < 8; ++v) {
        int r = m0 + v + 8 * khalf;
        float* po = out + (size_t)r * N_FEAT + n;
        po[0]  = acc0[v];
        po[16] = acc1[v];
        po[32] = acc2[v];
        po[48] = acc3[v];
    }
}

// ---------------- host-side orchestration ----------------

extern "C" void kernel_launch(void* const* d_in, const int* in_sizes, int n_in,
                              void* d_out, int out_size, void* d_ws, size_t ws_size,
                              hipStream_t stream) {
    const float* x       = (const float*)d_in[0];
    const int*   ei      = (const int*)d_in[1];        // [2, N_EDGES] flat
    const float* weights = (const float*)d_in[2];      // [3, 64, 64]
    const int* src = ei;
    const int* dst = ei + N_EDGES;
    float* out = (float*)d_out;

    // workspace: deg/inv_deg | agg | tmp   (~25.9 MB)
    float* deg = (float*)d_ws;
    float* agg = deg + 51200;                          // padded past N_NODES
    float* tmp = agg + (size_t)N_NODES * N_FEAT;

    // degree + inversion (once; shared across layers)
    init_deg_kernel<<<(N_NODES + 255) / 256, 256, 0, stream>>>(deg);
    count_deg_kernel<<<(N_EDGES + 255) / 256, 256, 0, stream>>>(dst, deg);
    invert_deg_kernel<<<(N_NODES + 255) / 256, 256, 0, stream>>>(deg);

    // layer chain: x -> d_out -> tmp -> d_out  (d_out doubles as scratch)
    const float* xc = x;
    float* outs[N_LAYERS] = { out, tmp, out };
    for (int l = 0; l < N_LAYERS; ++l) {
        self_copy_kernel<<<(N_NODES * N_FEAT / 4 + 255) / 256, 256, 0, stream>>>(xc, agg);
        scatter_kernel<<<(N_EDGES * 16 + 255) / 256, 256, 0, stream>>>(xc, src, dst, agg);
        gemm_wmma_kernel<<<(N_TILES + 3) / 4, 128, 0, stream>>>(
            agg, deg, weights + (size_t)l * N_FEAT * N_FEAT, outs[l]);
        xc = outs[l];
    }
}